// LIFNeuron_10522669875350
// MI455X (gfx1250) — compile-verified
//
#include <hip/hip_runtime.h>
#include <cstdint>
#include <cstddef>

// LIF scan: currents (T,B,DIM) f32 -> spikes (T,B,DIM) f32 ++ final membrane (B,DIM) f32
// T=64, B=256, DIM=1024. Pure streaming recurrence: memory-bound (~135 MB @ 23.3 TB/s).

typedef float v4f __attribute__((ext_vector_type(4)));

constexpr int T_STEPS = 64;
constexpr int BATCH   = 256;
constexpr int DIM     = 1024;
constexpr int BDIM    = BATCH * DIM;        // elements per timestep (262144)
constexpr int VEC     = 4;                  // floats per thread per step (16B)
constexpr int THREADS = 256;                // 8 wave32s per block
constexpr int TILE    = THREADS * VEC;      // 1024 floats = 4 KB per block-step
constexpr int NBLK    = BDIM / TILE;        // 256 blocks (one batch row each)
constexpr int STAGES  = 3;                  // triple-buffered LDS pipeline
constexpr int STAGE_BYTES = TILE * 4;       // 4096 B

// Async copy: 16B per lane, global -> LDS, tracked by ASYNCcnt.
// GVS form: mem = SADDR(64b sgpr pair) + VADDR(32b per-lane byte offset); VDST = LDS byte addr.
__device__ __forceinline__ void async_ld_b128(uint32_t lds_addr, uint32_t voff,
                                              const float* sbase) {
  asm volatile("global_load_async_to_lds_b128 %0, %1, %2"
               :: "v"(lds_addr), "v"(voff), "s"(sbase)
               : "memory");
}

__device__ __forceinline__ float lif_step(float cur, float beta, float astr,
                                          float& mem, float& ad) {
  const float thr  = 1.0f + ad;                                   // BASE_THRESHOLD + adaptation
  mem = __builtin_fmaf(beta, mem, cur);                           // beta*mem + cur
  const float hard = (mem >= thr) ? 1.0f : 0.0f;
  const float x    = mem - thr;
  const float sur  = x / __builtin_fmaf(25.0f, __builtin_fabsf(x), 1.0f);  // x/(1+25|x|)
  const float spk  = (hard - sur) + sur;   // forward value of straight-through spike
  mem = mem * (1.0f - spk);                // hard reset
  ad  = __builtin_fmaf(0.9f, ad, astr * spk);                     // ADAPT_DECAY*ad + a*spk
  return spk;
}

__global__ __launch_bounds__(THREADS, 2)
void lif_scan_kernel(const float* __restrict__ currents,
                     const float* __restrict__ beta_raw,
                     const float* __restrict__ astr_in,
                     float* __restrict__ out) {
  __shared__ float lds[STAGES * TILE];

  const int tid       = threadIdx.x;
  const int blockBase = blockIdx.x * TILE;                 // element offset within a timestep
  const uint32_t voff = (uint32_t)(tid * VEC * 4);         // per-lane byte offset (16B lanes)
  const uint32_t lbase = (uint32_t)(uintptr_t)(&lds[0]) + voff;
  const float* gbase  = currents + blockBase;

  // ---- prologue: stage async loads for t = 0, 1 ----
  async_ld_b128(lbase + 0u * STAGE_BYTES, voff, gbase);
  async_ld_b128(lbase + 1u * STAGE_BYTES, voff, gbase + (size_t)BDIM);

  // per-thread parameters (TILE == DIM, so d0 = tid*VEC, 16B aligned)
  const int d0 = (blockBase + tid * VEC) % DIM;
  const v4f br = *(const v4f*)(beta_raw + d0);
  const v4f as = *(const v4f*)(astr_in + d0);

  float beta[VEC], astr[VEC], mem[VEC], ad[VEC];
#pragma unroll
  for (int e = 0; e < VEC; ++e) {
    beta[e] = 1.0f / (1.0f + __expf(-br[e]));   // sigmoid(beta_raw)
    astr[e] = __builtin_fmaxf(as[e], 0.0f);     // relu(adaptation_strength)
    mem[e]  = 0.0f;
    ad[e]   = 0.0f;
  }

  // ---- main pipelined loop: compute t, keep t+1/t+2 in flight ----
#pragma unroll 3
  for (int t = 0; t < T_STEPS - 1; ++t) {
    // loads complete in order: <=1 outstanding ==> tile t has landed in LDS
    asm volatile("s_wait_asynccnt 0x1" ::: "memory");

    if (t + 2 < T_STEPS)
      async_ld_b128(lbase + (uint32_t)((t + 2) % STAGES) * STAGE_BYTES, voff,
                    gbase + (size_t)(t + 2) * BDIM);
    if (t + 8 < T_STEPS)  // warm L2 well ahead (global_prefetch_b8, no counter)
      __builtin_prefetch(gbase + (size_t)(t + 8) * BDIM + tid * VEC);

    const v4f cur = *(const v4f*)&lds[(t % STAGES) * TILE + tid * VEC];
    v4f spk;
#pragma unroll
    for (int e = 0; e < VEC; ++e)
      spk[e] = lif_step(cur[e], beta[e], astr[e], mem[e], ad[e]);

    __builtin_nontemporal_store(
        spk, (v4f*)(out + (size_t)t * BDIM + blockBase + tid * VEC));
  }

  // ---- peeled last step: require all async loads done ----
  asm volatile("s_wait_asynccnt 0x0" ::: "memory");
  {
    const int t = T_STEPS - 1;
    const v4f cur = *(const v4f*)&lds[(t % STAGES) * TILE + tid * VEC];
    v4f spk;
#pragma unroll
    for (int e = 0; e < VEC; ++e)
      spk[e] = lif_step(cur[e], beta[e], astr[e], mem[e], ad[e]);
    __builtin_nontemporal_store(
        spk, (v4f*)(out + (size_t)t * BDIM + blockBase + tid * VEC));
  }

  // ---- final membrane after the spikes block ----
  v4f m4;
#pragma unroll
  for (int e = 0; e < VEC; ++e) m4[e] = mem[e];
  __builtin_nontemporal_store(
      m4, (v4f*)(out + (size_t)T_STEPS * BDIM + blockBase + tid * VEC));
}

extern "C" void kernel_launch(void* const* d_in, const int* in_sizes, int n_in,
                              void* d_out, int out_size, void* d_ws, size_t ws_size,
                              hipStream_t stream) {
  const float* currents = (const float*)d_in[0];   // (T,B,DIM) f32
  const float* beta_raw = (const float*)d_in[1];   // (DIM,)    f32
  const float* astr     = (const float*)d_in[2];   // (DIM,)    f32
  float* out            = (float*)d_out;           // spikes ++ membrane

  (void)in_sizes; (void)n_in; (void)out_size; (void)d_ws; (void)ws_size;

  lif_scan_kernel<<<NBLK, THREADS, 0, stream>>>(currents, beta_raw, astr, out);
}